// SimpleGNNLayer_22454089023967
// MI455X (gfx1250) — compile-verified
//
#include <hip/hip_runtime.h>
#include <hip/hip_bf16.h>
#include <stdint.h>

typedef __attribute__((ext_vector_type(2))) float v2f;
typedef __attribute__((ext_vector_type(8))) float v8f;

#define IN_DIM  256
#define OUT_DIM 256

// ---------------------------------------------------------------------------
// Phase 0: zero the output accumulator and the degree array (d_out / d_ws are
// poisoned by the harness, so every call must re-initialize them).
// ---------------------------------------------------------------------------
__global__ __launch_bounds__(256) void gnn_zero_kernel(float* __restrict__ out,
                                                       float* __restrict__ deg,
                                                       int total_out, int n_nodes) {
    int i = blockIdx.x * blockDim.x + threadIdx.x;
    if (i < total_out) out[i] = 0.0f;
    if (i < n_nodes)   deg[i] = 0.0f;
}

// ---------------------------------------------------------------------------
// Phase 1: xt = x @ W^T + b using V_WMMA_F32_16X16X4_F32.
// One wave computes a 16(M) x 64(N) strip: 4 accumulators, A reused 4x.
// A 16x4 f32 fragment: lanes 0-15 -> M=lane, K={0,1}; lanes 16-31 -> K={2,3}.
// B 4x16 f32 fragment mirrors A with N in place of M; B[k][n] = W[n][k].
// C/D 16x16 f32: VGPR r -> M=r (lanes 0-15) / M=r+8 (lanes 16-31), N=lane&15.
// 4 consecutive waves in a block share the same 16-row x tile -> WGP$ reuse.
// ---------------------------------------------------------------------------
__global__ __launch_bounds__(256) void gnn_gemm_wmma_kernel(const float* __restrict__ x,
                                                            const float* __restrict__ W,
                                                            const float* __restrict__ bias,
                                                            float* __restrict__ xt,
                                                            int n_nodes) {
    const int wave = (blockIdx.x * blockDim.x + threadIdx.x) >> 5;  // global wave id
    const int lane = threadIdx.x & 31;

    const int mtile = wave >> 2;          // 16-row tile index
    const int ngrp  = (wave & 3) * 64;    // 64-wide column group
    const int m0    = mtile * 16;
    if (m0 >= n_nodes) return;

    const int row16 = lane & 15;                 // M (for A) / N (for B)
    const int koff  = (lane >> 4) << 1;          // 0 or 2: which K-pair this lane holds

    v8f acc0 = {}, acc1 = {}, acc2 = {}, acc3 = {};

    const float* xrow = x + (size_t)(m0 + row16) * IN_DIM;
    const float* w0   = W + (size_t)(ngrp +  0 + row16) * IN_DIM;
    const float* w1   = W + (size_t)(ngrp + 16 + row16) * IN_DIM;
    const float* w2   = W + (size_t)(ngrp + 32 + row16) * IN_DIM;
    const float* w3   = W + (size_t)(ngrp + 48 + row16) * IN_DIM;

    #pragma unroll 8
    for (int k0 = 0; k0 < IN_DIM; k0 += 4) {
        const int k = k0 + koff;
        v2f a, b0, b1, b2, b3;
        a.x  = xrow[k]; a.y  = xrow[k + 1];
        b0.x = w0[k];   b0.y = w0[k + 1];
        b1.x = w1[k];   b1.y = w1[k + 1];
        b2.x = w2[k];   b2.y = w2[k + 1];
        b3.x = w3[k];   b3.y = w3[k + 1];
        acc0 = __builtin_amdgcn_wmma_f32_16x16x4_f32(false, a, false, b0, (short)0, acc0, false, false);
        acc1 = __builtin_amdgcn_wmma_f32_16x16x4_f32(false, a, false, b1, (short)0, acc1, false, false);
        acc2 = __builtin_amdgcn_wmma_f32_16x16x4_f32(false, a, false, b2, (short)0, acc2, false, false);
        acc3 = __builtin_amdgcn_wmma_f32_16x16x4_f32(false, a, false, b3, (short)0, acc3, false, false);
    }

    // Store D + bias. Lane holds column (ngrp + t*16 + (lane&15)),
    // rows m0 + (lane<16 ? 0 : 8) + r for VGPR r.
    const int moff = (lane >> 4) << 3;   // 0 or 8
    const int ncol = lane & 15;
    v8f accs[4] = {acc0, acc1, acc2, acc3};
    #pragma unroll
    for (int t = 0; t < 4; ++t) {
        const int cg = ngrp + t * 16 + ncol;
        const float bv = bias[cg];
        float* dst = xt + (size_t)(m0 + moff) * OUT_DIM + cg;
        #pragma unroll
        for (int r = 0; r < 8; ++r) {
            dst[(size_t)r * OUT_DIM] = accs[t][r] + bv;
        }
    }
}

// ---------------------------------------------------------------------------
// Native no-return f32 atomic add (STOREcnt-tracked, TH[0]=0, resolves at L2).
// Inline asm guarantees global_atomic_add_f32 instead of a CAS-loop expansion.
// Device scope: edges targeting one node are processed on arbitrary WGPs.
// ---------------------------------------------------------------------------
__device__ __forceinline__ void atomic_add_f32(float* p, float v) {
    asm volatile("global_atomic_add_f32 %0, %1, off scope:SCOPE_DEV"
                 :: "v"((unsigned long long)(uintptr_t)p), "v"(v)
                 : "memory");
}

// ---------------------------------------------------------------------------
// Phase 2: edge gather + scatter-add. 64 lanes per edge, float4 gather from
// xt (L2-resident: 102 MB < 192 MB L2), native f32 atomic adds at L2.
// Lane with c==0 also bumps the in-degree counter.
// ---------------------------------------------------------------------------
__global__ __launch_bounds__(256) void gnn_scatter_kernel(const float* __restrict__ xt,
                                                          const int* __restrict__ src,
                                                          const int* __restrict__ tgt,
                                                          float* __restrict__ out,
                                                          float* __restrict__ deg,
                                                          int n_edges) {
    const long long id = (long long)blockIdx.x * blockDim.x + threadIdx.x;
    const int e = (int)(id >> 6);
    if (e >= n_edges) return;
    const int c = (int)(id & 63) << 2;

    const int s = src[e];
    const int t = tgt[e];

    const float4 v = *(const float4*)(xt + (size_t)s * OUT_DIM + c);
    float* o = out + (size_t)t * OUT_DIM + c;
    atomic_add_f32(o + 0, v.x);
    atomic_add_f32(o + 1, v.y);
    atomic_add_f32(o + 2, v.z);
    atomic_add_f32(o + 3, v.w);
    if (c == 0) atomic_add_f32(deg + t, 1.0f);
}

// ---------------------------------------------------------------------------
// Phase 3: out[n][:] /= (deg[n] + 1)
// ---------------------------------------------------------------------------
__global__ __launch_bounds__(256) void gnn_norm_kernel(float* __restrict__ out,
                                                       const float* __restrict__ deg,
                                                       int n_nodes) {
    const long long id = (long long)blockIdx.x * blockDim.x + threadIdx.x;
    const int node = (int)(id >> 6);
    if (node >= n_nodes) return;
    const int c = (int)(id & 63) << 2;

    const float inv = 1.0f / (deg[node] + 1.0f);
    float4* p = (float4*)(out + (size_t)node * OUT_DIM + c);
    float4 v = *p;
    v.x *= inv; v.y *= inv; v.z *= inv; v.w *= inv;
    *p = v;
}

// ---------------------------------------------------------------------------
// Launcher. d_ws layout: [xt: N*256 f32][deg: N f32]
// ---------------------------------------------------------------------------
extern "C" void kernel_launch(void* const* d_in, const int* in_sizes, int n_in,
                              void* d_out, int out_size, void* d_ws, size_t ws_size,
                              hipStream_t stream) {
    const float* x    = (const float*)d_in[0];
    const float* W    = (const float*)d_in[1];
    const float* bias = (const float*)d_in[2];
    const int*   ei   = (const int*)d_in[3];   // [2, E] row-major int32
    float* out = (float*)d_out;

    const int n_nodes = in_sizes[0] / IN_DIM;   // 100000
    const int n_edges = in_sizes[3] / 2;        // 1600000

    float* xt  = (float*)d_ws;
    float* deg = xt + (size_t)n_nodes * OUT_DIM;

    // Phase 0: zero out + deg
    const int total_out = n_nodes * OUT_DIM;
    {
        int blocks = (total_out + 255) / 256;
        gnn_zero_kernel<<<blocks, 256, 0, stream>>>(out, deg, total_out, n_nodes);
    }

    // Phase 1: WMMA GEMM. waves = (n_nodes/16) * 4, 8 waves per block.
    {
        const int mtiles = (n_nodes + 15) / 16;         // 6250
        const long long waves = (long long)mtiles * 4;  // 25000
        const long long threads = waves * 32;
        int blocks = (int)((threads + 255) / 256);
        gnn_gemm_wmma_kernel<<<blocks, 256, 0, stream>>>(x, W, bias, xt, n_nodes);
    }

    // Phase 2: scatter-add over edges (64 lanes / edge, float4 each)
    {
        const long long threads = (long long)n_edges * 64;
        int blocks = (int)((threads + 255) / 256);
        gnn_scatter_kernel<<<blocks, 256, 0, stream>>>(xt, ei, ei + n_edges, out, deg, n_edges);
    }

    // Phase 3: normalize
    {
        const long long threads = (long long)n_nodes * 64;
        int blocks = (int)((threads + 255) / 256);
        gnn_norm_kernel<<<blocks, 256, 0, stream>>>(out, deg, n_nodes);
    }
}